// GreedyCTCDecoder_16458314678384
// MI455X (gfx1250) — compile-verified
//
#include <hip/hip_runtime.h>
#include <stdint.h>

// Problem shape (reference: N=64, C=128, T=4096; N derived from in_sizes).
constexpr int C      = 128;
constexpr int T      = 4096;
constexpr int TW     = 64;    // t-columns per LDS tile
constexpr int TCHUNK = 512;   // t-columns per block (kernel 1)
constexpr int NTHR   = 256;   // block size

#if defined(__has_builtin)
# if __has_builtin(__builtin_amdgcn_tensor_load_to_lds)
#  define HAVE_TDM 1
# endif
#endif
#ifndef HAVE_TDM
# define HAVE_TDM 0
#endif

typedef unsigned int v4u __attribute__((ext_vector_type(4)));
typedef int          v8i __attribute__((ext_vector_type(8)));
typedef int          v4i __attribute__((ext_vector_type(4)));

// ---------------------------------------------------------------------------
// Path A: Tensor Data Mover (one descriptor per C x TW tile), TENSORcnt.
// Descriptor layout per CDNA5 ISA.
// This toolchain exposes the 6-arg builtin:
//   (v4u group0, v8i group1, v4i group2, v4i group3, v8i extra, i32 cpol)
// ---------------------------------------------------------------------------
#if HAVE_TDM
__device__ __forceinline__ void tdm_load_tile(unsigned lds_off, const float* gtile) {
    const uint64_t ga = (uint64_t)(uintptr_t)gtile;
    v4u g0;
    g0.x = 1u;                                            // count=1, user descriptor
    g0.y = lds_off;                                       // lds_addr (bytes)
    g0.z = (unsigned)(ga & 0xFFFFFFFFu);                  // global_addr[31:0]
    g0.w = (unsigned)((ga >> 32) & 0x1FFFFFFu)            // global_addr[56:32]
         | (2u << 30);                                    // type = 2 ("image")
    v8i g1;
    g1[0] = 0x20000;                                      // data_size=2 (4 B); mask/flags=0
    g1[1] = (int)(((unsigned)T & 0xFFFFu) << 16);         // tensor_dim0[15:0] (bits 63:48)
    g1[2] = (int)((((unsigned)T >> 16) & 0xFFFFu)         // tensor_dim0[31:16]
         | (((unsigned)C & 0xFFFFu) << 16));              // tensor_dim1[15:0]
    g1[3] = (int)((((unsigned)C >> 16) & 0xFFFFu)         // tensor_dim1[31:16]
         | ((unsigned)TW << 16));                         // tile_dim0 = TW
    g1[4] = C;                                            // tile_dim1 = C, tile_dim2 = 0
    g1[5] = T;                                            // tensor_dim0_stride[31:0] = T elems
    g1[6] = 0;                                            // stride hi / dim1_stride lo
    g1[7] = 0;                                            // dim1_stride hi
    const v4i z4 = {0, 0, 0, 0};                          // groups 2/3: 2-D tensor
    const v8i z8 = {0, 0, 0, 0, 0, 0, 0, 0};
    __builtin_amdgcn_tensor_load_to_lds(g0, g1, z4, z4, z8, 0);
}
#endif

// ---------------------------------------------------------------------------
// Path B (fallback): per-lane async global->LDS copies, ASYNCcnt.
// ---------------------------------------------------------------------------
__device__ __forceinline__ void async_load_b128(unsigned lds_byte_off, const float* gaddr) {
    asm volatile("global_load_async_to_lds_b128 %0, %1, off"
                 :: "v"(lds_byte_off), "v"(gaddr)
                 : "memory");
}

__device__ __forceinline__ void wait_asynccnt0() {
    asm volatile("s_wait_asynccnt 0x0" ::: "memory");
}

__device__ __forceinline__ void issue_tile_async(const float* base_nc, unsigned lds_base,
                                                 int t0, int tid) {
#pragma unroll
    for (int k = 0; k < 8; ++k) {
        int e  = tid + (k << 8);     // 0..2047 float4 index
        int c  = e >> 4;             // 16 float4 per row
        int j4 = e & 15;
        const float* g = base_nc + (size_t)c * T + t0 + (j4 << 2);
        async_load_b128(lds_base + (unsigned)(e << 4), g);
    }
}

// ---------------------------------------------------------------------------
// Kernel 1: pred[n,t] = argmax_c logits[n,c,t]  (HBM streaming, double buffer)
// ---------------------------------------------------------------------------
__global__ void __launch_bounds__(NTHR)
ctc_argmax_kernel(const float* __restrict__ logits, int* __restrict__ pred) {
    __shared__ float tile[2][C * TW];   // 2 x 32 KB double buffer
    __shared__ float pv[4 * TW];        // partial best values  (4 c-groups)
    __shared__ int   pc[4 * TW];        // partial best indices

    const int chunksPerSample = T / TCHUNK;                 // 8
    const int n       = blockIdx.x / chunksPerSample;
    const int t0chunk = (blockIdx.x % chunksPerSample) * TCHUNK;
    const int tid     = threadIdx.x;

    const float* base_nc = logits + (size_t)n * C * T;
    const int ntiles = TCHUNK / TW;                         // 8

    const unsigned lds_buf0 = (unsigned)(uintptr_t)&tile[0][0];
    const unsigned lds_buf1 = (unsigned)(uintptr_t)&tile[1][0];

#if HAVE_TDM
    const bool issuer = (tid < 32);                         // wave 0 drives the TDM
    if (issuer) {
        tdm_load_tile(lds_buf0, base_nc + t0chunk);
        __builtin_amdgcn_s_wait_tensorcnt(0);
    }
    __syncthreads();
#else
    issue_tile_async(base_nc, lds_buf0, t0chunk, tid);
    wait_asynccnt0();
    __syncthreads();
#endif

    for (int i = 0; i < ntiles; ++i) {
        // Prefetch next tile into the other buffer while reducing this one.
        if (i + 1 < ntiles) {
            const unsigned dst = ((i + 1) & 1) ? lds_buf1 : lds_buf0;
            const int tnext = t0chunk + (i + 1) * TW;
#if HAVE_TDM
            if (issuer) tdm_load_tile(dst, base_nc + tnext);
#else
            issue_tile_async(base_nc, dst, tnext, tid);
#endif
        }

        const float* tb = &tile[i & 1][0];
        const int t_local = tid & (TW - 1);
        const int cg      = tid >> 6;          // 0..3, each covers 32 classes
        const int c_first = cg << 5;

        // Ascending-c strictly-greater compare == first-max tie-break (jnp.argmax).
        float bv = tb[c_first * TW + t_local];
        int   bc = c_first;
#pragma unroll 4
        for (int c = c_first + 1; c < c_first + 32; ++c) {
            float v = tb[c * TW + t_local];
            if (v > bv) { bv = v; bc = c; }
        }
        pv[cg * TW + t_local] = bv;
        pc[cg * TW + t_local] = bc;
        __syncthreads();

        if (tid < TW) {
            float best = pv[tid];
            int   bcc  = pc[tid];
#pragma unroll
            for (int g = 1; g < 4; ++g) {
                float v = pv[g * TW + tid];
                if (v > best) { best = v; bcc = pc[g * TW + tid]; }
            }
            pred[(size_t)n * T + t0chunk + i * TW + tid] = bcc;
        }
        __syncthreads();                       // everyone done with buffer (i&1)

        if (i + 1 < ntiles) {
#if HAVE_TDM
            if (issuer) __builtin_amdgcn_s_wait_tensorcnt(0);
#else
            wait_asynccnt0();
#endif
            __syncthreads();                   // next tile published to all waves
        }
    }
}

// ---------------------------------------------------------------------------
// Kernel 2: collapse repeats, drop blanks, left-compact, lengths.
// One block per sample; 256 threads x 16 t each; block-wide scan.
// ---------------------------------------------------------------------------
__global__ void __launch_bounds__(NTHR)
ctc_compact_kernel(const int* __restrict__ pred, const int* __restrict__ blank_p,
                   int* __restrict__ decoded, int* __restrict__ lengths) {
    constexpr int PER = T / NTHR;              // 16
    __shared__ int s[NTHR];

    const int n   = blockIdx.x;
    const int tid = threadIdx.x;
    const int blank = *blank_p;

    const int* p   = pred + (size_t)n * T;
    const int base = tid * PER;

    int prev = (base == 0) ? -1 : p[base - 1];
    int v[PER];
#pragma unroll
    for (int j = 0; j < PER; ++j) v[j] = p[base + j];

    unsigned keepmask = 0;
    int cnt = 0;
#pragma unroll
    for (int j = 0; j < PER; ++j) {
        int cur = v[j];
        bool k = (cur != blank) && (cur != prev);
        keepmask |= (k ? 1u : 0u) << j;
        cnt += k ? 1 : 0;
        prev = cur;
    }

    // Inclusive Hillis-Steele scan of per-thread counts.
    s[tid] = cnt;
    __syncthreads();
    for (int d = 1; d < NTHR; d <<= 1) {
        int t = (tid >= d) ? s[tid - d] : 0;
        __syncthreads();
        s[tid] += t;
        __syncthreads();
    }
    int off   = s[tid] - cnt;                  // exclusive prefix
    int total = s[NTHR - 1];

    int* dout = decoded + (size_t)n * T;
#pragma unroll
    for (int j = 0; j < PER; ++j) dout[base + j] = -1;   // PAD
    __threadfence_block();
    __syncthreads();

#pragma unroll
    for (int j = 0; j < PER; ++j) {
        if ((keepmask >> j) & 1u) dout[off++] = v[j];
    }
    if (tid == 0) lengths[n] = total;
}

// ---------------------------------------------------------------------------
extern "C" void kernel_launch(void* const* d_in, const int* in_sizes, int n_in,
                              void* d_out, int out_size, void* d_ws, size_t ws_size,
                              hipStream_t stream) {
    const float* logits = (const float*)d_in[0];
    const int*   blank  = (const int*)d_in[1];

    const int N = in_sizes[0] / (C * T);       // 64

    int* pred    = (int*)d_ws;                 // N*T int32 scratch (1 MB)
    int* decoded = (int*)d_out;                // N*T int32
    int* lengths = decoded + (size_t)N * T;    // N int32

    ctc_argmax_kernel<<<N * (T / TCHUNK), NTHR, 0, stream>>>(logits, pred);
    ctc_compact_kernel<<<N, NTHR, 0, stream>>>(pred, blank, decoded, lengths);
}